// Binarization_70875550318898
// MI455X (gfx1250) — compile-verified
//
#include <hip/hip_runtime.h>
#include <hip/hip_bf16.h>
#include <stdint.h>

// Problem geometry (fixed by the reference)
#define ROWS  4096
#define COLS  8192
#define TPB   256
#define EPT   (COLS / TPB)     // 32 elements per thread
#define NWAVE (TPB / 32)       // 8 wave32 waves per block
#define NT    6                // params trajectory: t = 0..5
#define QMAX  3.0f
#define QEPS  1e-8f

#define AS1 __attribute__((address_space(1)))
#define AS3 __attribute__((address_space(3)))

typedef __attribute__((__vector_size__(16))) int v4i;

#if __has_builtin(__builtin_amdgcn_global_load_async_to_lds_b128)
#define HAVE_ASYNC_LDS 1
#pragma message("gfx1250 async global->LDS builtin: AVAILABLE")
#else
#define HAVE_ASYNC_LDS 0
#pragma message("gfx1250 async global->LDS builtin: MISSING (sync fallback)")
#endif

// One 16-byte global->LDS async copy (ASYNCcnt-tracked on gfx1250).
__device__ __forceinline__ void g2lds_b128(const void* g, void* l) {
#if HAVE_ASYNC_LDS
  // AS(1) value == generic value for global; AS(3) value == low 32 bits of
  // a generic LDS pointer (aperture lives in the high dword).
  __builtin_amdgcn_global_load_async_to_lds_b128(
      (AS1 v4i*)(uintptr_t)g, (AS3 v4i*)(uint32_t)(uintptr_t)l, 0, 0);
#else
  *(float4*)l = *(const float4*)g;
#endif
}

__device__ __forceinline__ void async_wait_all() {
#if HAVE_ASYNC_LDS
#if __has_builtin(__builtin_amdgcn_s_wait_asynccnt)
  __builtin_amdgcn_s_wait_asynccnt(0);
#else
  asm volatile("s_wait_asynccnt 0" ::: "memory");
#endif
#endif
}

// ---------------------------------------------------------------------------
// Kernel 1: one block per row. Async-stage the row (x: 32KB, mask: 8KB) into
// LDS coalesced, read back register-resident (32 floats + 32 mask bytes per
// thread), then run: stats pass + 6 fused (err + ALS-update) passes.
// Emits per-row {scale, offset, err} for t = 0..5 into workspace.
// ---------------------------------------------------------------------------
__global__ __launch_bounds__(TPB) void qrow_kernel(
    const float* __restrict__ x, const uint8_t* __restrict__ mask,
    float* __restrict__ ws_scale, float* __restrict__ ws_off,
    float* __restrict__ ws_err)
{
  __shared__ float    lds_x[COLS];       // 32 KB
  __shared__ uint32_t lds_m[COLS / 4];   //  8 KB
  __shared__ float    red[NWAVE][4];

  const int tid  = threadIdx.x;
  const int lane = tid & 31;
  const int wid  = tid >> 5;
  const int row  = blockIdx.x;

  const float*   grow = x    + (size_t)row * COLS;
  const uint8_t* mrow = mask + (size_t)row * COLS;

  // Stage row into LDS: lane-contiguous 16B chunks -> perfectly coalesced.
#pragma unroll
  for (int i = 0; i < (COLS * 4) / (TPB * 16); ++i) {   // 8 iters
    const int off = (i * TPB + tid) * 16;
    g2lds_b128((const char*)grow + off, (char*)lds_x + off);
  }
#pragma unroll
  for (int i = 0; i < COLS / (TPB * 16); ++i) {         // 2 iters
    const int off = (i * TPB + tid) * 16;
    g2lds_b128((const char*)mrow + off, (char*)lds_m + off);
  }
  async_wait_all();
  __syncthreads();

  // Register-resident slice: columns [tid*EPT, tid*EPT+EPT)
  float    xv[EPT];
  uint32_t mw[EPT / 4];
#pragma unroll
  for (int i = 0; i < EPT / 4; ++i)
    *(float4*)&xv[i * 4] = *(const float4*)&lds_x[tid * EPT + i * 4];
#pragma unroll
  for (int i = 0; i < EPT / 16; ++i)
    *(uint4*)&mw[i * 4] = *(const uint4*)&lds_m[tid * (EPT / 4) + i * 4];

  // ---- stats pass: valid count, sum(x*m), masked min/max -------------------
  float vcnt = 0.f, sx = 0.f;
  float xmn = __builtin_inff(), xmx = -__builtin_inff();
#pragma unroll
  for (int j = 0; j < EPT; ++j) {
    const bool  m  = ((mw[j >> 2] >> ((j & 3) * 8)) & 0xffu) != 0u;
    const float xj = xv[j];
    vcnt += m ? 1.f : 0.f;
    sx   += m ? xj  : 0.f;
    xmn = fminf(xmn, m ? xj :  __builtin_inff());
    xmx = fmaxf(xmx, m ? xj : -__builtin_inff());
  }
#pragma unroll
  for (int off = 16; off >= 1; off >>= 1) {
    vcnt += __shfl_xor(vcnt, off, 32);
    sx   += __shfl_xor(sx,   off, 32);
    xmn = fminf(xmn, __shfl_xor(xmn, off, 32));
    xmx = fmaxf(xmx, __shfl_xor(xmx, off, 32));
  }
  __syncthreads();
  if (lane == 0) { red[wid][0] = vcnt; red[wid][1] = sx; red[wid][2] = xmn; red[wid][3] = xmx; }
  __syncthreads();
  vcnt = 0.f; sx = 0.f; xmn = __builtin_inff(); xmx = -__builtin_inff();
#pragma unroll
  for (int w = 0; w < NWAVE; ++w) {
    vcnt += red[w][0]; sx += red[w][1];
    xmn = fminf(xmn, red[w][2]); xmx = fmaxf(xmx, red[w][3]);
  }

  // ---- init params (t = 0) -------------------------------------------------
  const bool hv = vcnt > 0.f;
  float a = hv ? xmn : -1.f;
  float b = hv ? xmx :  1.f;
  a = fminf(a, 0.f);
  b = fmaxf(b, 0.f);
  float sc = fmaxf((b - a) / QMAX, QEPS);
  const float zp = fminf(fmaxf(rintf(-a / sc), 0.f), QMAX); // rint == round-half-even == jnp.round
  float of = -sc * zp;
  const float vcl = fmaxf(vcnt, 1.f);

  // ---- fused trajectory passes: err(params_t) + sums -> params_{t+1} -------
  for (int t = 0; t < NT; ++t) {
    const float inv = 1.f / sc;   // row-uniform reciprocal
    float A = 0.f, B = 0.f, Sq = 0.f, E = 0.f;
#pragma unroll
    for (int j = 0; j < EPT; ++j) {
      const bool  m  = ((mw[j >> 2] >> ((j & 3) * 8)) & 0xffu) != 0u;
      const float xo = xv[j] - of;
      float q = fminf(fmaxf(rintf(xo * inv), 0.f), QMAX);
      q = m ? q : 0.f;
      A  = fmaf(q, q, A);     // sum q^2
      B  = fmaf(xo, q, B);    // sum (x-o)*q
      Sq += q;                // sum q
      const float d = xv[j] - fmaf(sc, q, of);
      E += m ? d * d : 0.f;   // masked squared error
    }
#pragma unroll
    for (int off = 16; off >= 1; off >>= 1) {
      A  += __shfl_xor(A,  off, 32);
      B  += __shfl_xor(B,  off, 32);
      Sq += __shfl_xor(Sq, off, 32);
      E  += __shfl_xor(E,  off, 32);
    }
    __syncthreads();
    if (lane == 0) { red[wid][0] = A; red[wid][1] = B; red[wid][2] = Sq; red[wid][3] = E; }
    __syncthreads();
    A = B = Sq = E = 0.f;
#pragma unroll
    for (int w = 0; w < NWAVE; ++w) { A += red[w][0]; B += red[w][1]; Sq += red[w][2]; E += red[w][3]; }

    if (tid == 0) {
      ws_scale[t * ROWS + row] = sc;
      ws_off  [t * ROWS + row] = of;
      ws_err  [t * ROWS + row] = E;
    }

    // ALS update -> params_{t+1} (all threads compute identically)
    float ns = (A > QEPS) ? (B / (A + QEPS)) : sc;
    ns = fmaxf(fabsf(ns), QEPS);
    const float no = (sx - ns * Sq) / vcl;
    sc = hv ? ns : sc;
    of = hv ? no : of;
  }
}

// ---------------------------------------------------------------------------
// Kernel 2: single block; reduce per-row err over rows for each t, then pick
// t* = index of the running minimum with strict-< tracking (matches reference:
// global scalar `better` broadcast to all rows).
// ---------------------------------------------------------------------------
__global__ __launch_bounds__(TPB) void select_kernel(
    const float* __restrict__ ws_err, int* __restrict__ t_star)
{
  __shared__ float red[NWAVE];
  __shared__ float totals[NT];
  const int tid = threadIdx.x, lane = tid & 31, wid = tid >> 5;

  for (int t = 0; t < NT; ++t) {
    float s = 0.f;
    for (int r = tid; r < ROWS; r += TPB) s += ws_err[t * ROWS + r];
#pragma unroll
    for (int off = 16; off >= 1; off >>= 1) s += __shfl_xor(s, off, 32);
    __syncthreads();
    if (lane == 0) red[wid] = s;
    __syncthreads();
    if (tid == 0) {
      float tot = 0.f;
      for (int w = 0; w < NWAVE; ++w) tot += red[w];
      totals[t] = tot;
    }
  }
  __syncthreads();
  if (tid == 0) {
    float best = totals[0];
    int ts = 0;
    for (int t = 1; t < NT; ++t)
      if (totals[t] < best) { best = totals[t]; ts = t; }
    *t_star = ts;
  }
}

// ---------------------------------------------------------------------------
// Kernel 3: streaming requantize with the winning params; out = m ? xhat : x.
// 4 floats / thread, whole block stays inside one row (1024 | 8192).
// ---------------------------------------------------------------------------
__global__ __launch_bounds__(TPB) void writeback_kernel(
    const float* __restrict__ x, const uint8_t* __restrict__ mask,
    const float* __restrict__ ws_scale, const float* __restrict__ ws_off,
    const int* __restrict__ t_star, float* __restrict__ out)
{
  const int    ts   = *t_star;
  const size_t base = ((size_t)blockIdx.x * TPB + threadIdx.x) * 4;
  const int    row  = (int)(base >> 13);  // COLS == 8192

  const float sc  = ws_scale[(size_t)ts * ROWS + row];
  const float of  = ws_off  [(size_t)ts * ROWS + row];
  const float inv = 1.f / sc;

  const float4   xv = *(const float4*)(x + base);
  const uint32_t mv = *(const uint32_t*)(mask + base);
  const float xs[4] = {xv.x, xv.y, xv.z, xv.w};
  float4 o;
  float  os[4];
#pragma unroll
  for (int j = 0; j < 4; ++j) {
    const bool m = ((mv >> (j * 8)) & 0xffu) != 0u;
    float q = fminf(fmaxf(rintf((xs[j] - of) * inv), 0.f), QMAX);
    q = m ? q : 0.f;
    const float xh = fmaf(sc, q, of);
    os[j] = m ? xh : xs[j];
  }
  o.x = os[0]; o.y = os[1]; o.z = os[2]; o.w = os[3];
  *(float4*)(out + base) = o;
}

// ---------------------------------------------------------------------------
extern "C" void kernel_launch(void* const* d_in, const int* in_sizes, int n_in,
                              void* d_out, int out_size, void* d_ws, size_t ws_size,
                              hipStream_t stream) {
  (void)in_sizes; (void)n_in; (void)out_size; (void)ws_size;
  const float*   x    = (const float*)d_in[0];
  const uint8_t* mask = (const uint8_t*)d_in[1];   // jnp bool -> 1 byte/elem

  float* ws_scale = (float*)d_ws;                  // [NT][ROWS]
  float* ws_off   = ws_scale + (size_t)NT * ROWS;  // [NT][ROWS]
  float* ws_err   = ws_off   + (size_t)NT * ROWS;  // [NT][ROWS]
  int*   t_star   = (int*)(ws_err + (size_t)NT * ROWS);

  qrow_kernel<<<ROWS, TPB, 0, stream>>>(x, mask, ws_scale, ws_off, ws_err);
  select_kernel<<<1, TPB, 0, stream>>>(ws_err, t_star);
  writeback_kernel<<<(ROWS * (size_t)COLS) / (TPB * 4), TPB, 0, stream>>>(
      x, mask, ws_scale, ws_off, t_star, (float*)d_out);
}